// ReservoirCPC_54769422958724
// MI455X (gfx1250) — compile-verified
//
#include <hip/hip_runtime.h>
#include <math.h>

// Problem dims (from the reference)
#define IDIM 256   // INPUT_DIM
#define RES  1024  // RES_SIZE
#define LAT  128   // LATENT_DIM
#define BSZ  32    // batch
#define TT   1024  // time steps
// ALPHA = 0.9 -> h = 0.1*h + 0.9*tanh(...)

typedef __attribute__((ext_vector_type(2))) float        v2f;
typedef __attribute__((ext_vector_type(8))) float        v8f;
typedef __attribute__((ext_vector_type(4))) unsigned int v4u;
typedef __attribute__((ext_vector_type(8))) int          v8i;
typedef __attribute__((ext_vector_type(4))) int          v4i;

// CDNA5 fp32 WMMA: D(16x16,f32) = A(16x4,f32) * B(4x16,f32) + C
// A frag per lane: row m = lane&15, K pair kb = 2*(lane>>4)  -> {A[m][kb], A[m][kb+1]}
// B frag per lane: col n = lane&15, K pair kb = 2*(lane>>4)  -> {B[kb][n], B[kb+1][n]}
// C/D element i of lane L: m = i + 8*(L>>4), n = L&15
__device__ __forceinline__ v8f wmma_f32(v2f a, v2f b, v8f c) {
  return __builtin_amdgcn_wmma_f32_16x16x4_f32(
      /*neg_a=*/false, a, /*neg_b=*/false, b,
      /*c_mod=*/(short)0, c, /*reuse_a=*/false, /*reuse_b=*/false);
}

// ---------------------------------------------------------------------------
// Kernel A: U[row, n] = sum_k X[row, k] * Win[k, n]    (row = b*T + t)
// M = 32768, K = 256, N = 1024.  One wave per 16x16 tile, 8 waves/block.
// ---------------------------------------------------------------------------
__global__ void __launch_bounds__(256)
kA_xWin(const float* __restrict__ X, const float* __restrict__ Win,
        float* __restrict__ U) {
  const int lane = threadIdx.x & 31;
  const int wave = threadIdx.x >> 5;
  const int tile = blockIdx.x * 8 + wave;   // 131072 tiles = 2048 x 64
  const int mt = tile >> 6;                 // 0..2047
  const int nt = tile & 63;                 // 0..63
  const int m0 = mt * 16, n0 = nt * 16;
  const int ln = lane & 15;
  const int half = lane >> 4;
  const int kb = half * 2;

  const float* xrow = X + (long)(m0 + ln) * IDIM;
  const int n = n0 + ln;

  v8f acc = {};
  for (int k = 0; k < IDIM; k += 4) {
    v2f a = *(const v2f*)(xrow + k + kb);                 // 8B aligned (k+kb even)
    v2f b;
    b.x = Win[(long)(k + kb) * RES + n];
    b.y = Win[(long)(k + kb + 1) * RES + n];
    acc = wmma_f32(a, b, acc);
  }
#pragma unroll
  for (int i = 0; i < 8; ++i) {
    const int row = m0 + i + 8 * half;
    U[(long)row * RES + n] = acc[i];
  }
}

// ---------------------------------------------------------------------------
// Kernel B: persistent scan.  128 blocks x 128 threads (4 waves).
// Block owns one 16x16 tile of h (mt in {0,1}, nt in 0..63).
// W[:, n0:n0+16] (64KB) is staged into LDS ONCE via the Tensor Data Mover
// (2D tile descriptor, TENSORcnt tracked), then reused for all 1024 steps.
// Waves split K=1024 into 4 ranges (64 chained WMMAs each), LDS-reduce,
// wave 0 applies u + tanh + leaky blend, writes h_next and hs (aliased on U).
// Grid barrier: monotone phase counter in device memory.
// ---------------------------------------------------------------------------
__global__ void __launch_bounds__(128)
kB_scan(const float* __restrict__ W, float* __restrict__ UH,  // U in, hs out (same buffer)
        float* __restrict__ hA, float* __restrict__ hB,
        unsigned* __restrict__ cnt) {
  extern __shared__ float smem[];           // [0,16384): W tile, [16384,17408): reduce
  float* Wl  = smem;                        // Wl[k*16 + j] = W[k][n0+j]
  float* red = smem + RES * 16;

  const int tid  = threadIdx.x;
  const int lane = tid & 31;
  const int wave = tid >> 5;                // 0..3
  const int mt = blockIdx.x >> 6;           // 0..1
  const int nt = blockIdx.x & 63;           // 0..63
  const int m0 = mt * 16, n0 = nt * 16;
  const int ln = lane & 15;
  const int half = lane >> 4;
  const int kb = half * 2;

#if __has_builtin(__builtin_amdgcn_tensor_load_to_lds) && __has_builtin(__builtin_amdgcn_s_wait_tensorcnt)
  // --- TDM staging: DMA the 2D tile W[0:1024, n0:n0+16] into LDS offset 0 ---
  if (wave == 0) {                          // TDM issues once per wave; gate to wave 0
    const unsigned long long ga = (unsigned long long)(const void*)(W + n0);
    v4u g0;
    g0[0] = 1u;                                             // count=1, user mode, no gather
    g0[1] = 0u;                                             // lds_addr: Wl at LDS offset 0
    g0[2] = (unsigned)(ga & 0xFFFFFFFFu);                   // global_addr[31:0]
    g0[3] = (unsigned)((ga >> 32) & 0x1FFFFFFu) | (2u << 30); // global_addr[56:32] | type=2
    v8i g1;
    g1[0] = (int)(2u << 16);                                // data_size=2 (4 bytes)
    g1[1] = (int)((RES & 0xFFFF) << 16);                    // tensor_dim0 = 1024 (lo16)
    g1[2] = (int)(((RES >> 16) & 0xFFFF) | ((RES & 0xFFFF) << 16)); // dim0 hi | tensor_dim1 lo
    g1[3] = (int)(((RES >> 16) & 0xFFFF) | (16u << 16));    // tensor_dim1 hi | tile_dim0 = 16
    g1[4] = (int)(RES & 0xFFFF);                            // tile_dim1 = 1024, tile_dim2 = 0
    g1[5] = (int)RES;                                       // tensor_dim0_stride = 1024 (lo32)
    g1[6] = 0;                                              // stride hi | dim1_stride lo
    g1[7] = 0;
    const v4i gz4 = {0, 0, 0, 0};
    const v8i gz8 = {0, 0, 0, 0, 0, 0, 0, 0};
    __builtin_amdgcn_tensor_load_to_lds(g0, g1, gz4, gz4, gz8, 0);
    __builtin_amdgcn_s_wait_tensorcnt(0);
  }
#else
  // Fallback: manual staging of the 16-column W slice into LDS.
  for (int idx = tid; idx < RES * 16; idx += 128) {
    const int kk = idx >> 4, j = idx & 15;
    Wl[idx] = W[(long)kk * RES + n0 + j];
  }
#endif
  __syncthreads();

  const int kstart = wave * (RES / 4);      // 4-way K split
  const int kend   = kstart + (RES / 4);
  const unsigned nb = gridDim.x;
  unsigned phase = 0;

  for (int t = 0; t < TT; ++t) {
    const float* hc = (t & 1) ? hB : hA;    // ping-pong
    float*       hn = (t & 1) ? hA : hB;

    // partial (h @ W) for this K range
    v8f acc = {};
    const float* hrow = hc + (long)(m0 + ln) * RES;
    for (int k = kstart; k < kend; k += 4) {
      v2f a = *(const v2f*)(hrow + k + kb);
      v2f b;
      b.x = Wl[(k + kb) * 16 + ln];
      b.y = Wl[(k + kb + 1) * 16 + ln];
      acc = wmma_f32(a, b, acc);
    }
#pragma unroll
    for (int i = 0; i < 8; ++i) red[(wave * 8 + i) * 32 + lane] = acc[i];
    __syncthreads();

    if (wave == 0) {
#pragma unroll
      for (int i = 0; i < 8; ++i) {
        const float s = red[(0 * 8 + i) * 32 + lane] + red[(1 * 8 + i) * 32 + lane] +
                        red[(2 * 8 + i) * 32 + lane] + red[(3 * 8 + i) * 32 + lane];
        const int mrow = m0 + i + 8 * half;       // batch index
        const int ncol = n0 + ln;                 // reservoir unit
        const long uidx = (long)mrow * TT * RES + (long)t * RES + ncol;
        const float u    = UH[uidx];
        const float hold = hc[(long)mrow * RES + ncol];
        const float hv   = 0.1f * hold + 0.9f * tanhf(u + s);
        hn[(long)mrow * RES + ncol] = hv;
        UH[uidx] = hv;                            // hs[b,t,r] overwrites consumed u
        if (t + 1 < TT) __builtin_prefetch(&UH[uidx + RES], 0, 0);  // next step's u
      }
    }

    // device-wide barrier before the next step reads h_next
    __threadfence();
    __syncthreads();
    if (tid == 0) {
      phase += nb;
      __hip_atomic_fetch_add(cnt, 1u, __ATOMIC_ACQ_REL, __HIP_MEMORY_SCOPE_AGENT);
      while (__hip_atomic_load(cnt, __ATOMIC_ACQUIRE, __HIP_MEMORY_SCOPE_AGENT) < phase) {
        __builtin_amdgcn_s_sleep(1);
      }
    }
    __syncthreads();
  }
}

// ---------------------------------------------------------------------------
// Kernel C: dual-head projection.  z = hs@Wz^T + bz, c = hs@Wc^T + bc.
// M = 32768, K = 1024, N = 128.  One wave per tile; A-fragment (hs) is loaded
// once and feeds both accumulators.
// ---------------------------------------------------------------------------
__global__ void __launch_bounds__(256)
kC_proj(const float* __restrict__ HS,
        const float* __restrict__ Wz, const float* __restrict__ bz,
        const float* __restrict__ Wc, const float* __restrict__ bc,
        float* __restrict__ Z, float* __restrict__ C) {
  const int lane = threadIdx.x & 31;
  const int wave = threadIdx.x >> 5;
  const int tile = blockIdx.x * 8 + wave;   // 16384 tiles = 2048 x 8
  const int mt = tile >> 3;
  const int nt = tile & 7;
  const int m0 = mt * 16, n0 = nt * 16;
  const int ln = lane & 15;
  const int half = lane >> 4;
  const int kb = half * 2;

  const float* hrow  = HS + (long)(m0 + ln) * RES;
  const float* wzrow = Wz + (long)(n0 + ln) * RES;  // B[k][n] = Wz[n][k] (contig in k)
  const float* wcrow = Wc + (long)(n0 + ln) * RES;

  v8f az = {}, ac = {};
  for (int k = 0; k < RES; k += 4) {
    v2f a   = *(const v2f*)(hrow + k + kb);
    v2f bzv = *(const v2f*)(wzrow + k + kb);
    v2f bcv = *(const v2f*)(wcrow + k + kb);
    az = wmma_f32(a, bzv, az);
    ac = wmma_f32(a, bcv, ac);
  }
  const float zb = bz[n0 + ln];
  const float cb = bc[n0 + ln];
#pragma unroll
  for (int i = 0; i < 8; ++i) {
    const int row = m0 + i + 8 * half;
    Z[(long)row * LAT + n0 + ln] = az[i] + zb;
    C[(long)row * LAT + n0 + ln] = ac[i] + cb;
  }
}

// ---------------------------------------------------------------------------
extern "C" void kernel_launch(void* const* d_in, const int* in_sizes, int n_in,
                              void* d_out, int out_size, void* d_ws, size_t ws_size,
                              hipStream_t stream) {
  const float* x   = (const float*)d_in[0];  // [32,1024,256]
  const float* Win = (const float*)d_in[1];  // [256,1024]
  const float* W   = (const float*)d_in[2];  // [1024,1024]
  const float* Wz  = (const float*)d_in[3];  // [128,1024]
  const float* bz  = (const float*)d_in[4];  // [128]
  const float* Wc  = (const float*)d_in[5];  // [128,1024]
  const float* bc  = (const float*)d_in[6];  // [128]

  float* z = (float*)d_out;                          // [32,1024,128]
  float* c = z + (long)BSZ * TT * LAT;               // [32,1024,128]

  // Workspace layout: [ U / hs : 128MB ][ hA : 128KB ][ hB : 128KB ][ counter ]
  char* ws = (char*)d_ws;
  const size_t UB = (size_t)BSZ * TT * RES * sizeof(float);   // 134217728
  const size_t HB = (size_t)BSZ * RES * sizeof(float);        // 131072
  float*    U   = (float*)ws;
  float*    hA  = (float*)(ws + UB);
  float*    hBp = (float*)(ws + UB + HB);
  unsigned* cnt = (unsigned*)(ws + UB + 2 * HB);

  // zero h0 (ping-pong buffers) and the barrier counter
  (void)hipMemsetAsync(ws + UB, 0, 2 * HB + 64, stream);

  // 1) U = X @ W_in
  kA_xWin<<<16384, 256, 0, stream>>>(x, Win, U);

  // 2) persistent scan (68KB dynamic LDS: 64KB W slice + 4KB reduction)
  const size_t shmB = (RES * 16 + 4 * 8 * 32) * sizeof(float);
  (void)hipFuncSetAttribute((const void*)kB_scan,
                            hipFuncAttributeMaxDynamicSharedMemorySize, (int)shmB);
  kB_scan<<<128, 128, shmB, stream>>>(W, U, hA, hBp, cnt);

  // 3) z, c heads (reads hs once, two WMMA accumulators)
  kC_proj<<<2048, 256, 0, stream>>>(U, Wz, bz, Wc, bc, z, c);
}